// AdaptiveTokenSampler_3307124818206
// MI455X (gfx1250) — compile-verified
//
#include <hip/hip_runtime.h>
#include <hip/hip_bf16.h>
#include <math.h>

typedef float vf2 __attribute__((ext_vector_type(2)));
typedef float vf8 __attribute__((ext_vector_type(8)));

#define NTOK 197
#define BIMG 64
#define CH   768
#define NH   12
#define HD   64
#define TSEL 128
#define NN   (NTOK*NTOK)   // 38809
#define VPAD 80            // LDS row stride (dwords) for conflict-free dual-half reads

__device__ __forceinline__ vf8 wmma4(vf2 a, vf2 b, vf8 c) {
  // V_WMMA_F32_16X16X4_F32 : D = A(16x4) x B(4x16) + C, fp32 throughout
  return __builtin_amdgcn_wmma_f32_16x16x4_f32(false, a, false, b, (short)0, c, false, false);
}

__device__ __forceinline__ float wred_max(float v) {
#pragma unroll
  for (int o = 16; o > 0; o >>= 1) v = fmaxf(v, __shfl_xor(v, o, 32));
  return v;
}
__device__ __forceinline__ float wred_sum(float v) {
#pragma unroll
  for (int o = 16; o > 0; o >>= 1) v += __shfl_xor(v, o, 32);
  return v;
}

// ---------------- K1: qkv = x @ w_qkv + b, scatter to q/k/v with policy ----
// Wave tile: 32(M) x 64(N) -> 8 accumulators, B-loads amortized over 8 WMMAs.
__global__ __launch_bounds__(128) void k_qkv(
    const float* __restrict__ x, const float* __restrict__ w,
    const float* __restrict__ bias, const float* __restrict__ policy,
    float* __restrict__ q, float* __restrict__ k, float* __restrict__ v)
{
  const int wave = threadIdx.x >> 5;
  const int lane = threadIdx.x & 31;
  const int half = lane >> 4;
  const int l    = lane & 15;
  const int m0 = blockIdx.x * 32;                 // M = 12608 = 394*32
  const int j0 = (blockIdx.y * 4 + wave) * 64;    // 2304 = 9 * (4 waves * 64)

  const float* __restrict__ ar0 = x + (size_t)(m0 + l) * CH;
  const float* __restrict__ ar1 = x + (size_t)(m0 + 16 + l) * CH;

  vf8 acc[2][4];
#pragma unroll
  for (int mi = 0; mi < 2; mi++)
#pragma unroll
    for (int t = 0; t < 4; t++)
#pragma unroll
      for (int r = 0; r < 8; r++) acc[mi][t][r] = 0.f;

  for (int k0 = 0; k0 < CH; k0 += 4) {
    const int ka = k0 + 2 * half;
    vf2 a0, a1;
    a0.x = ar0[ka]; a0.y = ar0[ka + 1];
    a1.x = ar1[ka]; a1.y = ar1[ka + 1];
    const float* b0 = w + (size_t)ka * (3 * CH) + j0 + l;
    const float* b1 = b0 + 3 * CH;
#pragma unroll
    for (int t = 0; t < 4; t++) {
      vf2 bb; bb.x = b0[t * 16]; bb.y = b1[t * 16];
      acc[0][t] = wmma4(a0, bb, acc[0][t]);
      acc[1][t] = wmma4(a1, bb, acc[1][t]);
    }
  }
#pragma unroll
  for (int mi = 0; mi < 2; mi++) {
#pragma unroll
    for (int t = 0; t < 4; t++) {
#pragma unroll
      for (int r = 0; r < 8; r++) {
        const int m = m0 + mi * 16 + r + 8 * half;
        const int n_tok = m >> 6;          // row = n*B + b, B = 64
        const int b_img = m & 63;
        const float p = policy[b_img * NTOK + n_tok];
        const int j = j0 + t * 16 + l;
        const int which = j / CH;
        const int rem = j - which * CH;
        const int h = rem >> 6;
        const int d = rem & 63;
        const float val = (acc[mi][t][r] + bias[j]) * p;
        float* dst = (which == 0) ? q : (which == 1) ? k : v;
        dst[((size_t)(b_img * NH + h) * NTOK + n_tok) * HD + d] = val;
      }
    }
  }
}

// ---------------- K2: attn = q k^T * scale, per (b,h) ----------------------
__global__ __launch_bounds__(416) void k_scores(
    const float* __restrict__ q, const float* __restrict__ kk, float* __restrict__ attn)
{
  const int wave = threadIdx.x >> 5;   // 13 waves: one n-tile each
  const int lane = threadIdx.x & 31;
  const int half = lane >> 4;
  const int l    = lane & 15;
  const int bh = blockIdx.y;
  const int m0 = blockIdx.x * 16;
  const int n0 = wave * 16;
  const int mrow = m0 + l, nrow = n0 + l;
  const bool mok = mrow < NTOK, nok = nrow < NTOK;
  const float* __restrict__ ar = q  + ((size_t)bh * NTOK + (mok ? mrow : 0)) * HD;
  const float* __restrict__ br = kk + ((size_t)bh * NTOK + (nok ? nrow : 0)) * HD;

  vf8 c;
#pragma unroll
  for (int r = 0; r < 8; r++) c[r] = 0.f;

#pragma unroll
  for (int k0 = 0; k0 < HD; k0 += 4) {
    const int ka = k0 + 2 * half;
    vf2 a, b;
    a.x = mok ? ar[ka]     : 0.f;
    a.y = mok ? ar[ka + 1] : 0.f;
    b.x = nok ? br[ka]     : 0.f;   // B[k][n] = K[n][k] -> contiguous
    b.y = nok ? br[ka + 1] : 0.f;
    c = wmma4(a, b, c);
  }
  float* out = attn + (size_t)bh * NN;
#pragma unroll
  for (int r = 0; r < 8; r++) {
    const int m = m0 + r + 8 * half;
    const int n = n0 + l;
    if (m < NTOK && n < NTOK) out[m * NTOK + n] = c[r] * 0.125f;  // hd^-0.5
  }
}

// ---------------- K3: softmax_with_policy, one wave per row ----------------
__global__ __launch_bounds__(256) void k_softmax(
    float* __restrict__ attn, const float* __restrict__ policy)
{
  const int wave = threadIdx.x >> 5;
  const int lane = threadIdx.x & 31;
  const int row = blockIdx.x * 8 + wave;   // B*H*N = 151296 = 18912*8
  const int bh = row / NTOK;
  const int i  = row - bh * NTOK;
  const int b  = bh / NH;
  float* a = attn + (size_t)bh * NN + (size_t)i * NTOK;
  const float* p = policy + b * NTOK;

  float mx = -3.402823e38f;
  for (int n = lane; n < NTOK; n += 32) mx = fmaxf(mx, a[n]);
  mx = wred_max(mx);

  float s = 0.f;
  for (int n = lane; n < NTOK; n += 32) {
    const float pn = p[n];
    const float ap = pn + (1.f - pn) * ((n == i) ? 1.f : 0.f);
    const float e = expf(a[n] - mx) * ap;
    a[n] = e;
    s += e;
  }
  s = wred_sum(s);
  const float eN  = 1e-6f / (float)NTOK;
  const float inv = 1.f / (s + 1e-6f);
  for (int n = lane; n < NTOK; n += 32) a[n] = (a[n] + eN) * inv;
}

// ---------------- K4: v_norm[b,n] = || v[b,:,n,:] ||_2 ---------------------
__global__ __launch_bounds__(256) void k_vnorm(const float* __restrict__ v, float* __restrict__ vn)
{
  const int wave = threadIdx.x >> 5;
  const int lane = threadIdx.x & 31;
  const int id = blockIdx.x * 8 + wave;   // B*N = 12608 = 1576*8
  const int b = id / NTOK, n = id - b * NTOK;
  const float* base = v + ((size_t)(b * NH) * NTOK + n) * HD;
  float s = 0.f;
#pragma unroll
  for (int h = 0; h < NH; h++) {
    const float* pr = base + (size_t)h * NTOK * HD;
    for (int d = lane; d < HD; d += 32) { const float t = pr[d]; s += t * t; }
  }
  s = wred_sum(s);
  if (lane == 0) vn[id] = sqrtf(s);
}

// ---------------- K5: significance scores, one block per batch -------------
__global__ __launch_bounds__(256) void k_sig(
    const float* __restrict__ attn, const float* __restrict__ vn, float* __restrict__ sig)
{
  const int b = blockIdx.x, tid = threadIdx.x;
  __shared__ float red[256];
  float val = 0.f;
  if (tid < NTOK - 1) {
    const int n = 1 + tid;
    float s = 0.f;
#pragma unroll
    for (int h = 0; h < NH; h++) s += attn[(size_t)(b * NH + h) * NN + n];  // row 0
    val = s * vn[b * NTOK + n];
  }
  red[tid] = val;
  __syncthreads();
  for (int o = 128; o > 0; o >>= 1) { if (tid < o) red[tid] += red[tid + o]; __syncthreads(); }
  const float tot = red[0];
  if (tid < NTOK - 1) sig[b * (NTOK - 1) + tid] = val / tot;
}

// ---------------- K6: sort/CDF/pick/uniq -> pol + gather map ---------------
__global__ __launch_bounds__(256) void k_pick(
    const float* __restrict__ sig, float* __restrict__ pol_out, int* __restrict__ tok_src)
{
  const int b = blockIdx.x, tid = threadIdx.x;
  __shared__ float s_sig[196], s_sorted[196], s_ncdf[196];
  __shared__ int   s_order[196], s_a[127], s_b[127];
  __shared__ float s_ystart;

  for (int i = tid; i < 196; i += 256) s_sig[i] = sig[b * 196 + i];
  __syncthreads();
  // stable ascending rank sort (matches jnp.argsort stable)
  for (int i = tid; i < 196; i += 256) {
    const float si = s_sig[i];
    int r = 0;
    for (int j = 0; j < 196; j++) {
      const float sj = s_sig[j];
      r += (sj < si) || (sj == si && j < i);
    }
    s_order[r] = i; s_sorted[r] = si;
  }
  __syncthreads();
  if (tid == 0) {
    float acc = 0.f;
    for (int i = 0; i < 196; i++) { acc += s_sorted[i]; s_sorted[i] = acc; }  // cdf in place
    float cmin = 3.4e38f, cmax = -3.4e38f;
    for (int i = 0; i < 196; i++) { cmin = fminf(cmin, s_sorted[i]); cmax = fmaxf(cmax, s_sorted[i]); }
    float ms = 3.4e38f;
    for (int i = 0; i < 196; i++) {
      const float nv = (s_sorted[i] - cmin) / (cmax - cmin);
      s_ncdf[i] = nv;
      const float t = nv + ((nv == 0.f) ? 1e8f : 0.f);
      ms = fminf(ms, t);
    }
    s_ystart = ms;
  }
  __syncthreads();
  const float ys_start = s_ystart;
  // pick: R-1=127 targets, diff_tokens = -69 -> padded = ncdf[69:196]
  for (int t = tid; t < 127; t += 256) {
    const float ylin = (float)t / 126.f;
    const float ys = ys_start + (ylin * 126.f - ys_start * (float)t) / 126.f;
    float best = 3.4e38f; int bi = 0;
    for (int idx = 0; idx < 127; idx++) {
      const float d = fabsf(ys - s_ncdf[69 + idx]);
      if (d < best) { best = d; bi = idx; }   // first-min like jnp.argmin
    }
    s_a[t] = bi + 69;
  }
  __syncthreads();
  for (int t = tid; t < 127; t += 256) {       // sort pick -> s
    const int vv = s_a[t]; int r = 0;
    for (int j = 0; j < 127; j++) { const int u = s_a[j]; r += (u < vv) || (u == vv && j < t); }
    s_b[r] = vv;
  }
  __syncthreads();
  for (int t = tid; t < 127; t += 256) {       // mark duplicates -> 196
    const int sl = (t < 126) ? s_b[t + 1] : 1;
    s_a[t] = (sl == s_b[t]) ? 196 : s_b[t];
  }
  __syncthreads();
  for (int t = tid; t < 127; t += 256) {       // sort uniq
    const int vv = s_a[t]; int r = 0;
    for (int j = 0; j < 127; j++) { const int u = s_a[j]; r += (u < vv) || (u == vv && j < t); }
    s_b[r] = vv;
  }
  __syncthreads();
  for (int t = tid; t < TSEL; t += 256) {
    float pv; int src;
    if (t == 0) { pv = 1.f; src = 0; }
    else {
      const int u = s_b[t - 1];
      if (u != 196) { pv = 1.f; src = 1 + s_order[u]; }
      else          { pv = 0.f; src = -1; }
    }
    pol_out[b * TSEL + t] = pv;
    tok_src[b * TSEL + t] = src;
  }
}

// ---------------- K9: selected_x gather ------------------------------------
__global__ __launch_bounds__(256) void k_select(
    const float* __restrict__ raw_x, const int* __restrict__ tok_src, float* __restrict__ out_sel)
{
  const int row = blockIdx.x;            // B*T = 8192
  const int b = row >> 7;
  const int src = tok_src[row];
  float* o = out_sel + (size_t)row * CH;
  if (src >= 0) {
    const float* r = raw_x + ((size_t)src * BIMG + b) * CH;
    for (int c = threadIdx.x; c < CH; c += 256) o[c] = r[c];
  } else {
    for (int c = threadIdx.x; c < CH; c += 256) o[c] = 0.f;
  }
}

// ---------------- K7: xo_pre = gathered(attn) @ v --------------------------
// V tile (197x64, 50KB) is shared by all 8 waves -> stage once in LDS.
// Row stride 80 dwords: the +2-row read of lanes 16..31 lands +160 dwords
// (bank +32), giving the two 16-lane halves disjoint bank ranges.
__global__ __launch_bounds__(256) void k_av(
    const float* __restrict__ attn, const float* __restrict__ v,
    const int* __restrict__ tok_src, float* __restrict__ xo)
{
  __shared__ float vs[NTOK * VPAD];      // 63,040 B of 320KB/WGP
  const int tid  = threadIdx.x;
  const int wave = tid >> 5;             // 8 waves cover M = 128
  const int lane = tid & 31;
  const int half = lane >> 4;
  const int l    = lane & 15;
  const int bh = blockIdx.x;
  const int b = bh / NH, h = bh - b * NH;
  const float* __restrict__ vb = v + (size_t)bh * NTOK * HD;

  // cooperative staged copy, float4 granularity (HD%4==0 keeps rows intact)
  for (int i4 = tid; i4 < (NTOK * HD) / 4; i4 += 256) {
    const int e  = i4 * 4;
    const int kk = e >> 6;        // row
    const int d  = e & 63;        // col
    const float4 val = *(const float4*)(vb + e);
    *(float4*)(vs + kk * VPAD + d) = val;
  }
  __syncthreads();

  const int m0 = wave * 16;
  const int trow = m0 + l;
  const int src = tok_src[b * TSEL + trow];
  const bool has = src >= 0;
  const float* __restrict__ ar = attn + (size_t)bh * NN + (size_t)(has ? src : 0) * NTOK;

  vf8 acc[4];
#pragma unroll
  for (int t = 0; t < 4; t++)
#pragma unroll
    for (int r = 0; r < 8; r++) acc[t][r] = 0.f;

  for (int k0 = 0; k0 < NTOK; k0 += 4) {
    const int ka = k0 + 2 * half;
    const bool k0ok = ka < NTOK, k1ok = (ka + 1) < NTOK;
    vf2 a;
    a.x = (has && k0ok) ? ar[ka]     : 0.f;
    a.y = (has && k1ok) ? ar[ka + 1] : 0.f;
#pragma unroll
    for (int t = 0; t < 4; t++) {
      vf2 bb;
      bb.x = k0ok ? vs[ka * VPAD + t * 16 + l]       : 0.f;
      bb.y = k1ok ? vs[(ka + 1) * VPAD + t * 16 + l] : 0.f;
      acc[t] = wmma4(a, bb, acc[t]);
    }
  }
#pragma unroll
  for (int t = 0; t < 4; t++)
#pragma unroll
    for (int r = 0; r < 8; r++) {
      const int row = m0 + r + 8 * half;
      xo[((size_t)b * TSEL + row) * CH + h * HD + t * 16 + l] = acc[t][r];
    }
}

// ---------------- K8: out = (xo_pre @ w_proj + b) * pol --------------------
// Wave tile: 32(M) x 64(N) like k_qkv.
__global__ __launch_bounds__(128) void k_proj(
    const float* __restrict__ xo, const float* __restrict__ w,
    const float* __restrict__ bias, const float* __restrict__ pol,
    float* __restrict__ out)
{
  const int wave = threadIdx.x >> 5;
  const int lane = threadIdx.x & 31;
  const int half = lane >> 4;
  const int l    = lane & 15;
  const int m0 = blockIdx.x * 32;                 // M = 8192 = 256*32
  const int j0 = (blockIdx.y * 4 + wave) * 64;    // 768 = 3 * (4 waves * 64)
  const float* __restrict__ ar0 = xo + (size_t)(m0 + l) * CH;
  const float* __restrict__ ar1 = xo + (size_t)(m0 + 16 + l) * CH;

  vf8 acc[2][4];
#pragma unroll
  for (int mi = 0; mi < 2; mi++)
#pragma unroll
    for (int t = 0; t < 4; t++)
#pragma unroll
      for (int r = 0; r < 8; r++) acc[mi][t][r] = 0.f;

  for (int k0 = 0; k0 < CH; k0 += 4) {
    const int ka = k0 + 2 * half;
    vf2 a0, a1;
    a0.x = ar0[ka]; a0.y = ar0[ka + 1];
    a1.x = ar1[ka]; a1.y = ar1[ka + 1];
    const float* b0 = w + (size_t)ka * CH + j0 + l;
    const float* b1 = b0 + CH;
#pragma unroll
    for (int t = 0; t < 4; t++) {
      vf2 bb; bb.x = b0[t * 16]; bb.y = b1[t * 16];
      acc[0][t] = wmma4(a0, bb, acc[0][t]);
      acc[1][t] = wmma4(a1, bb, acc[1][t]);
    }
  }
#pragma unroll
  for (int mi = 0; mi < 2; mi++)
#pragma unroll
    for (int t = 0; t < 4; t++)
#pragma unroll
      for (int r = 0; r < 8; r++) {
        const int m = m0 + mi * 16 + r + 8 * half;
        const int j = j0 + t * 16 + l;
        out[(size_t)m * CH + j] = (acc[mi][t][r] + bias[j]) * pol[m];
      }
}

extern "C" void kernel_launch(void* const* d_in, const int* in_sizes, int n_in,
                              void* d_out, int out_size, void* d_ws, size_t ws_size,
                              hipStream_t stream)
{
  const float* x      = (const float*)d_in[0];
  const float* policy = (const float*)d_in[1];
  const float* raw_x  = (const float*)d_in[2];
  // d_in[3] = sampler (unused)
  const float* w_qkv  = (const float*)d_in[4];
  const float* b_qkv  = (const float*)d_in[5];
  const float* w_proj = (const float*)d_in[6];
  const float* b_proj = (const float*)d_in[7];
  // d_in[8]=n_tokens(128), d_in[9]=n_ref_tokens(128): compile-time constants here

  const size_t QS = (size_t)BIMG * NH * NTOK * HD;  // 9,682,944 floats
  const size_t AS = (size_t)BIMG * NH * NN;         // 29,805,312 floats
  float* q    = (float*)d_ws;
  float* kk   = q + QS;
  float* v    = kk + QS;
  float* attn = v + QS;
  float* vn   = attn + AS;                          // B*N
  float* sig  = vn + (size_t)BIMG * NTOK;           // B*196
  float* xo   = sig + (size_t)BIMG * 196;           // B*T*C
  int* tok_src = (int*)(xo + (size_t)BIMG * TSEL * CH);

  float* out_xo  = (float*)d_out;                       // (B,T,C)
  float* out_sel = out_xo + (size_t)BIMG * TSEL * CH;   // (B,T,C)
  float* out_pol = out_sel + (size_t)BIMG * TSEL * CH;  // (B,T,1)

  k_qkv    <<<dim3(394, 9),  128, 0, stream>>>(x, w_qkv, b_qkv, policy, q, kk, v);
  k_scores <<<dim3(13, 768), 416, 0, stream>>>(q, kk, attn);
  k_softmax<<<18912,         256, 0, stream>>>(attn, policy);
  k_vnorm  <<<1576,          256, 0, stream>>>(v, vn);
  k_sig    <<<64,            256, 0, stream>>>(attn, vn, sig);
  k_pick   <<<64,            256, 0, stream>>>(sig, out_pol, tok_src);
  k_select <<<8192,          256, 0, stream>>>(raw_x, tok_src, out_sel);
  k_av     <<<768,           256, 0, stream>>>(attn, v, tok_src, xo);
  k_proj   <<<dim3(512, 3),  128, 0, stream>>>(xo, w_proj, b_proj, out_pol, out_xo);
}